// LinearAttention_79826262163840
// MI455X (gfx1250) — compile-verified
//
#include <hip/hip_runtime.h>

// ---------------------------------------------------------------------------
// LinearAttention for MI455X (gfx1250), fp32 end-to-end via V_WMMA_F32_16X16X4_F32
// B=8, L=4096, D=512, H=8, C=64
// Data movement: double-buffered GLOBAL_LOAD_ASYNC_TO_LDS_B128 staging for the
// GEMMs, TDM tensor_load_to_lds (6-arg form) for the ctx/q tiles.
// Workspace: qkv (B,1536,L) 192MB | ctx (B,H,64,64) 1MB | out (B,512,L) 64MB
// ---------------------------------------------------------------------------

typedef float v2f __attribute__((ext_vector_type(2)));
typedef float v4f __attribute__((ext_vector_type(4)));
typedef float v8f __attribute__((ext_vector_type(8)));
typedef int   i32x4 __attribute__((ext_vector_type(4)));
typedef int   i32x8 __attribute__((ext_vector_type(8)));
typedef unsigned int u32x4 __attribute__((ext_vector_type(4)));

// address-space-qualified int4 for the async-copy builtin (param0 printed by
// clang as 'int4 __device__*' i.e. addrspace(1); LDS side is addrspace(3))
typedef __attribute__((address_space(1))) i32x4 gas_i32x4;
typedef __attribute__((address_space(3))) i32x4 las_i32x4;

#define HEADS 8
#define B_    8
#define L_    4096
#define D_    512
#define HID   512
#define QKVR  1536
#define KC    32           // K-slab staged in LDS per pipeline stage
#define LSTR  36           // LDS row stride: 32 + 4 pad floats (float4-aligned)
#define SPLITK 16

// ---- CDNA5 feature detection ----------------------------------------------
#if defined(__has_builtin)
#  if __has_builtin(__builtin_amdgcn_global_load_async_to_lds_b128) && \
      __has_builtin(__builtin_amdgcn_s_wait_asynccnt)
#    define ASYNC_LDS 1
#  endif
#  if __has_builtin(__builtin_amdgcn_tensor_load_to_lds) && \
      __has_builtin(__builtin_amdgcn_s_wait_tensorcnt)
#    define HAVE_TDM 1
#  endif
#endif

// ---- 16B global->LDS copy: async (VGPR-bypassing, ASYNCcnt) or sync --------
__device__ __forceinline__ void cp_b128(float* ldsDst, const float* gSrc) {
#ifdef ASYNC_LDS
  __builtin_amdgcn_global_load_async_to_lds_b128(
      (gas_i32x4*)gSrc, (las_i32x4*)ldsDst, 0, 0);
#else
  *(v4f*)ldsDst = *(const v4f*)gSrc;
#endif
}

__device__ __forceinline__ void wait_stage() {
#ifdef ASYNC_LDS
  __builtin_amdgcn_s_wait_asynccnt(0);
#endif
  __syncthreads();
}

// ---- TDM 2D tile load descriptor (D# groups 0/1 per CDNA5 ISA) ----------
#ifdef HAVE_TDM
__device__ __forceinline__ unsigned lds_off(const float* p) {
  return (unsigned)(unsigned long long)(__attribute__((address_space(3))) const float*)p;
}

// tileW x tileH elements (4B each), global row stride rowStride elems,
// optional LDS padding: padAmount(code) DWORDs after every padInterval(code) DWORDs
__device__ __forceinline__ void tdm_load_2d(const float* gsrc, unsigned ldsAddr,
                                            unsigned tileW, unsigned tileH,
                                            unsigned rowStride,
                                            unsigned tensorW, unsigned tensorH,
                                            unsigned padInterval, unsigned padAmount,
                                            bool padEn) {
  unsigned long long ga = (unsigned long long)gsrc;
  u32x4 g0;
  g0.x = 1u;                                                   // count=1 (valid D#)
  g0.y = ldsAddr;                                              // lds_addr
  g0.z = (unsigned)(ga & 0xFFFFFFFFu);                         // global_addr[31:0]
  g0.w = (unsigned)((ga >> 32) & 0x01FFFFFFu) | (2u << 30);    // addr[56:32]|type=2
  i32x8 g1;
  g1[0] = (int)((2u << 16) |                                   // data_size=4B
                (padEn ? (1u << 20) : 0u) |
                (padInterval << 22) | (padAmount << 25));
  g1[1] = (int)((tensorW & 0xFFFFu) << 16);                    // tensor_dim0 lo16
  g1[2] = (int)(((tensorW >> 16) & 0xFFFFu) | ((tensorH & 0xFFFFu) << 16));
  g1[3] = (int)(((tensorH >> 16) & 0xFFFFu) | ((tileW & 0xFFFFu) << 16));
  g1[4] = (int)(tileH & 0xFFFFu);                              // tile_dim1 (dim2=0)
  g1[5] = (int)rowStride;                                      // dim0_stride[31:0]
  g1[6] = 0;
  g1[7] = 0;
  i32x4 z4 = {0, 0, 0, 0};
  i32x8 z8 = {0, 0, 0, 0, 0, 0, 0, 0};
  __builtin_amdgcn_tensor_load_to_lds(g0, g1, z4, z4, z8, 0);  // 6-arg form
}
#endif

// ---- WMMA wrapper: D = A(16x4) * B(4x16) + C, full fp32 --------------------
__device__ __forceinline__ v8f wmma4(v2f a, v2f b, v8f c) {
  return __builtin_amdgcn_wmma_f32_16x16x4_f32(false, a, false, b, (short)0, c,
                                               false, false);
}

// ---- Fragment loaders ------------------------------------------------------
// A stored [m][k] (stride LSTR) or B stored [n][k]
__device__ __forceinline__ v2f frag_ab(const float* s, int row0, int k0) {
  int lane = threadIdx.x & 31;
  int idx  = lane & 15;
  int kb   = (lane >> 4) << 1;
  const float* p = s + (row0 + idx) * LSTR + k0 + kb;
  v2f r; r.x = p[0]; r.y = p[1];
  return r;
}
// Matrix stored [k][row] with arbitrary stride
__device__ __forceinline__ v2f frag_kn(const float* s, int stride, int row0, int k0) {
  int lane = threadIdx.x & 31;
  int idx  = lane & 15;
  int kb   = (lane >> 4) << 1;
  const float* p = s + (k0 + kb) * stride + row0 + idx;
  v2f r; r.x = p[0]; r.y = p[stride];
  return r;
}

__device__ __forceinline__ void acc_store(float* C, long ld, long m0, long n0,
                                          v8f a0, v8f a1) {
  int lane = threadIdx.x & 31;
  int idx  = lane & 15;
  int mh   = (lane >> 4) << 3;
#pragma unroll
  for (int r = 0; r < 8; ++r) {
    long m = m0 + mh + r;
    C[m * ld + n0 + idx]      = a0[r];
    C[m * ld + n0 + 16 + idx] = a1[r];
  }
}

// ===========================================================================
// Kernel 1: qkv[b][o][l] = sum_d Wqkv[o][d] * x[b][l][d]
// M=1536, N=4096, K=512; 64x64 tile/block; double-buffered async LDS staging
// ===========================================================================
__global__ __launch_bounds__(256) void qkv_gemm(const float* __restrict__ x,
                                                const float* __restrict__ Wqkv,
                                                float* __restrict__ qkv) {
  __shared__ float As[2][64 * LSTR];
  __shared__ float Bs[2][64 * LSTR];
  int  b     = blockIdx.z;
  long mBase = (long)blockIdx.y * 64;
  long nBase = (long)blockIdx.x * 64;
  const float* xb = x + (long)b * L_ * D_;
  float* outb     = qkv + (long)b * QKVR * L_;

  int t  = threadIdx.x;
  int tr = t >> 3;             // 0..31
  int tc = (t & 7) << 2;       // 0..28
  auto stage = [&](int buf, int k) {
    cp_b128(&As[buf][tr * LSTR + tc],        Wqkv + (mBase + tr) * D_ + k + tc);
    cp_b128(&As[buf][(tr + 32) * LSTR + tc], Wqkv + (mBase + tr + 32) * D_ + k + tc);
    cp_b128(&Bs[buf][tr * LSTR + tc],        xb + (nBase + tr) * D_ + k + tc);
    cp_b128(&Bs[buf][(tr + 32) * LSTR + tc], xb + (nBase + tr + 32) * D_ + k + tc);
  };

  int w  = t >> 5;
  int m0 = (w & 3) << 4;
  int n0 = (w >> 2) << 5;
  v8f acc0 = {}; v8f acc1 = {};

  stage(0, 0);
  wait_stage();
  int cur = 0;
  for (int k = 0; k < D_; k += KC) {
    if (k + KC < D_) stage(cur ^ 1, k + KC);
#ifndef ASYNC_LDS
    if (k + KC < D_)
      __builtin_prefetch((const void*)(xb + (nBase + tr) * D_ + k + 2 * KC), 0, 1);
#endif
#pragma unroll
    for (int kk = 0; kk < KC; kk += 4) {
      v2f a  = frag_ab(As[cur], m0, kk);
      v2f b0 = frag_ab(Bs[cur], n0, kk);
      v2f b1 = frag_ab(Bs[cur], n0 + 16, kk);
      acc0 = wmma4(a, b0, acc0);
      acc1 = wmma4(a, b1, acc1);
    }
    wait_stage();
    cur ^= 1;
  }
  acc_store(outb, L_, mBase + m0, nBase + n0, acc0, acc1);
}

// ===========================================================================
// Kernel 2: softmax over L for k-rows (o in [512,1024)), in place
// ===========================================================================
__global__ __launch_bounds__(256) void softmax_rows(float* __restrict__ qkv) {
  long row = blockIdx.x;
  long b   = row >> 9;
  long o   = 512 + (row & 511);
  float* p = qkv + (b * (long)QKVR + o) * (long)L_;
  __shared__ float red[8];
  __shared__ float bc;
  int t = threadIdx.x, w = t >> 5, lane = t & 31;

  float mx = -3.4e38f;
  for (int i = t; i < L_; i += 256) mx = fmaxf(mx, p[i]);
#pragma unroll
  for (int off = 16; off; off >>= 1) mx = fmaxf(mx, __shfl_xor(mx, off, 32));
  if (lane == 0) red[w] = mx;
  __syncthreads();
  if (t == 0) {
    float m = red[0];
#pragma unroll
    for (int i = 1; i < 8; ++i) m = fmaxf(m, red[i]);
    bc = m;
  }
  __syncthreads();
  mx = bc;

  float s = 0.f;
  for (int i = t; i < L_; i += 256) s += expf(p[i] - mx);
#pragma unroll
  for (int off = 16; off; off >>= 1) s += __shfl_xor(s, off, 32);
  if (lane == 0) red[w] = s;
  __syncthreads();
  if (t == 0) {
    float sm = 0.f;
#pragma unroll
    for (int i = 0; i < 8; ++i) sm += red[i];
    bc = 1.0f / sm;
  }
  __syncthreads();
  float inv = bc;
  for (int i = t; i < L_; i += 256) p[i] = expf(p[i] - mx) * inv;
}

// ===========================================================================
// Kernel 3: softmax over C=64 for q (o in [0,512)), per (b,h,l), in place
// ===========================================================================
__global__ __launch_bounds__(256) void softmax_cols(float* __restrict__ qkv) {
  long tid = (long)blockIdx.x * 256 + threadIdx.x;
  long l   = tid & (L_ - 1);
  long bh  = tid >> 12;
  long b   = bh >> 3;
  long h   = bh & 7;
  float* base = qkv + (b * (long)QKVR + h * 64) * (long)L_ + l;
  float v[64];
  float mx = -3.4e38f;
#pragma unroll
  for (int c = 0; c < 64; ++c) { v[c] = base[c * (long)L_]; mx = fmaxf(mx, v[c]); }
  float s = 0.f;
#pragma unroll
  for (int c = 0; c < 64; ++c) { v[c] = expf(v[c] - mx); s += v[c]; }
  float inv = 1.0f / s;
#pragma unroll
  for (int c = 0; c < 64; ++c) base[c * (long)L_] = v[c] * inv;
}

// ===========================================================================
// Kernel 4a: zero ctx buffer
// ===========================================================================
__global__ __launch_bounds__(256) void zero_buf(float* __restrict__ p, long n) {
  long i = (long)blockIdx.x * 256 + threadIdx.x;
  if (i < n) p[i] = 0.f;
}

// ===========================================================================
// Kernel 4: ctx[b][h][d][e] = sum_n k[d][n]*v[e][n]   split-K + atomicAdd f32
// ===========================================================================
__global__ __launch_bounds__(256) void context_gemm(const float* __restrict__ qkv,
                                                    float* __restrict__ ctx) {
  __shared__ float As[2][64 * LSTR];
  __shared__ float Bs[2][64 * LSTR];
  int split = blockIdx.x;
  int h = blockIdx.y, b = blockIdx.z;
  const float* kmat = qkv + ((long)b * QKVR + 512  + h * 64) * (long)L_;
  const float* vmat = qkv + ((long)b * QKVR + 1024 + h * 64) * (long)L_;
  float* cmat = ctx + ((long)b * HEADS + h) * 64 * 64;

  int t  = threadIdx.x;
  int tr = t >> 3;
  int tc = (t & 7) << 2;
  auto stage = [&](int buf, long k) {
    cp_b128(&As[buf][tr * LSTR + tc],        kmat + (long)tr * L_ + k + tc);
    cp_b128(&As[buf][(tr + 32) * LSTR + tc], kmat + (long)(tr + 32) * L_ + k + tc);
    cp_b128(&Bs[buf][tr * LSTR + tc],        vmat + (long)tr * L_ + k + tc);
    cp_b128(&Bs[buf][(tr + 32) * LSTR + tc], vmat + (long)(tr + 32) * L_ + k + tc);
  };

  int w  = t >> 5;
  int m0 = (w & 3) << 4;
  int n0 = (w >> 2) << 5;
  v8f acc0 = {}; v8f acc1 = {};
  long kStart = (long)split * (L_ / SPLITK);
  long kEnd   = kStart + L_ / SPLITK;

  stage(0, kStart);
  wait_stage();
  int cur = 0;
  for (long k = kStart; k < kEnd; k += KC) {
    if (k + KC < kEnd) stage(cur ^ 1, k + KC);
#pragma unroll
    for (int kk = 0; kk < KC; kk += 4) {
      v2f a  = frag_ab(As[cur], m0, kk);
      v2f b0 = frag_ab(Bs[cur], n0, kk);
      v2f b1 = frag_ab(Bs[cur], n0 + 16, kk);
      acc0 = wmma4(a, b0, acc0);
      acc1 = wmma4(a, b1, acc1);
    }
    wait_stage();
    cur ^= 1;
  }
  int lane = threadIdx.x & 31;
  int idx  = lane & 15;
  int mh   = (lane >> 4) << 3;
#pragma unroll
  for (int r = 0; r < 8; ++r) {
    int m = m0 + mh + r;
    atomicAdd(&cmat[m * 64 + n0 + idx],      acc0[r]);
    atomicAdd(&cmat[m * 64 + n0 + 16 + idx], acc1[r]);
  }
}

// ===========================================================================
// Kernel 5: out[b][h*64+e][n] = sum_d ctx[d][e]*q[d][n]
// ctx + q tiles fetched by the Tensor Data Mover (one wave issues, all wait)
// ===========================================================================
__global__ __launch_bounds__(256) void apply_ctx(const float* __restrict__ qkv,
                                                 const float* __restrict__ ctx,
                                                 float* __restrict__ outbuf) {
  __shared__ float Cs[64 * 64];   // [d][e]
  __shared__ float Qs[64 * 68];   // [d][n], stride 68 = 64 + 4-DWORD TDM pad
  long nBase = (long)blockIdx.x * 64;
  int h = blockIdx.y, b = blockIdx.z;
  const float* qmat = qkv + ((long)b * QKVR + h * 64) * (long)L_;
  const float* cmat = ctx + ((long)b * HEADS + h) * 4096;
  float* omat = outbuf + ((long)b * HID + h * 64) * (long)L_;

#ifdef HAVE_TDM
  if (threadIdx.x < 32) {  // wave 0 drives the DMA engine
    tdm_load_2d(cmat, lds_off(Cs), 64, 64, 64, 64, 64, 0, 0, false);
    tdm_load_2d(qmat + nBase, lds_off(Qs), 64, 64, L_, L_, 64,
                /*padInterval=64dw*/5, /*padAmount=4dw*/3, true);
    __builtin_amdgcn_s_wait_tensorcnt(0);
  }
  __syncthreads();
#else
  {
    int t = threadIdx.x;
#pragma unroll
    for (int i = 0; i < 4; ++i) {                     // ctx: 4096 floats
      int id = t + i * 256;
      *(v4f*)(Cs + id * 4) = *(const v4f*)(cmat + id * 4);
    }
#pragma unroll
    for (int i = 0; i < 4; ++i) {                     // q tile: 64x64
      int id = t + i * 256;
      int r = id >> 4, c4 = (id & 15) << 2;
      *(v4f*)(Qs + r * 68 + c4) = *(const v4f*)(qmat + (long)r * L_ + nBase + c4);
    }
  }
  __syncthreads();
#endif

  int w  = threadIdx.x >> 5;
  int m0 = (w & 3) << 4;    // e
  int n0 = (w >> 2) << 5;   // n
  v8f acc0 = {}; v8f acc1 = {};
#pragma unroll
  for (int kk = 0; kk < 64; kk += 4) {
    v2f a  = frag_kn(Cs, 64, m0, kk);        // A[e][d] = Cs[d][e]
    v2f b0 = frag_kn(Qs, 68, n0, kk);        // B[d][n] = Qs[d][n]
    v2f b1 = frag_kn(Qs, 68, n0 + 16, kk);
    acc0 = wmma4(a, b0, acc0);
    acc1 = wmma4(a, b1, acc1);
  }
  acc_store(omat, L_, m0, nBase + n0, acc0, acc1);
}

// ===========================================================================
// Kernel 6: y[b][l][d] = sum_c out[b][c][l]*Wout[d][c] + bout[d]
// M=4096(l), N=512(d), K=512(c); double-buffered async LDS staging
// ===========================================================================
__global__ __launch_bounds__(256) void out_proj(const float* __restrict__ outbuf,
                                                const float* __restrict__ Wout,
                                                const float* __restrict__ bout,
                                                float* __restrict__ y) {
  __shared__ float Asb[2][KC * 68];    // [k=c][m=l], stride 68
  __shared__ float Bsb[2][64 * LSTR];  // [n=d][k=c]
  long lBase = (long)blockIdx.x * 64;
  long dBase = (long)blockIdx.y * 64;
  int  b = blockIdx.z;
  const float* ob = outbuf + (long)b * HID * L_;
  float* yb = y + (long)b * L_ * D_;

  int t   = threadIdx.x;
  int tr  = t >> 3;             // 0..31 (B tile rows)
  int tc  = (t & 7) << 2;
  int tr2 = t >> 4;             // 0..15 (A tile rows, 2 passes)
  int tc2 = (t & 15) << 2;
  auto stage = [&](int buf, int k) {
    cp_b128(&Asb[buf][tr2 * 68 + tc2],        ob + (long)(k + tr2) * L_ + lBase + tc2);
    cp_b128(&Asb[buf][(tr2 + 16) * 68 + tc2], ob + (long)(k + tr2 + 16) * L_ + lBase + tc2);
    cp_b128(&Bsb[buf][tr * LSTR + tc],        Wout + (dBase + tr) * HID + k + tc);
    cp_b128(&Bsb[buf][(tr + 32) * LSTR + tc], Wout + (dBase + tr + 32) * HID + k + tc);
  };

  int w  = t >> 5;
  int m0 = (w & 3) << 4;
  int n0 = (w >> 2) << 5;
  v8f acc0 = {}; v8f acc1 = {};

  stage(0, 0);
  wait_stage();
  int cur = 0;
  for (int k = 0; k < HID; k += KC) {
    if (k + KC < HID) stage(cur ^ 1, k + KC);
#pragma unroll
    for (int kk = 0; kk < KC; kk += 4) {
      v2f a  = frag_kn(Asb[cur], 68, m0, kk);
      v2f b0 = frag_ab(Bsb[cur], n0, kk);
      v2f b1 = frag_ab(Bsb[cur], n0 + 16, kk);
      acc0 = wmma4(a, b0, acc0);
      acc1 = wmma4(a, b1, acc1);
    }
    wait_stage();
    cur ^= 1;
  }
  int lane = threadIdx.x & 31;
  int idx  = lane & 15;
  int mh   = (lane >> 4) << 3;
#pragma unroll
  for (int r = 0; r < 8; ++r) {
    long l  = lBase + m0 + mh + r;
    long d0 = dBase + n0 + idx;
    yb[l * D_ + d0]      = acc0[r] + bout[d0];
    yb[l * D_ + d0 + 16] = acc1[r] + bout[d0 + 16];
  }
}

// ===========================================================================
extern "C" void kernel_launch(void* const* d_in, const int* in_sizes, int n_in,
                              void* d_out, int out_size, void* d_ws, size_t ws_size,
                              hipStream_t stream) {
  const float* x    = (const float*)d_in[0];
  const float* Wqkv = (const float*)d_in[1];
  const float* Wout = (const float*)d_in[2];
  const float* bout = (const float*)d_in[3];
  float* y = (float*)d_out;

  float* qkv = (float*)d_ws;
  float* ctx = qkv + (long)B_ * QKVR * L_;
  float* obf = ctx + (long)B_ * HEADS * 64 * 64;

  qkv_gemm    <<<dim3(L_ / 64, QKVR / 64, B_), 256, 0, stream>>>(x, Wqkv, qkv);
  softmax_rows<<<B_ * HID, 256, 0, stream>>>(qkv);
  softmax_cols<<<(B_ * HEADS * L_) / 256, 256, 0, stream>>>(qkv);
  zero_buf    <<<(B_ * HEADS * 64 * 64) / 256, 256, 0, stream>>>(ctx, (long)B_ * HEADS * 64 * 64);
  context_gemm<<<dim3(SPLITK, HEADS, B_), 256, 0, stream>>>(qkv, ctx);
  apply_ctx   <<<dim3(L_ / 64, HEADS, B_), 256, 0, stream>>>(qkv, ctx, obf);
  out_proj    <<<dim3(L_ / 64, D_ / 64, B_), 256, 0, stream>>>(obf, Wout, bout, y);
}